// GCN2_5970004541987
// MI455X (gfx1250) — compile-verified
//
#include <hip/hip_runtime.h>
#include <hip/hip_bf16.h>

typedef __attribute__((ext_vector_type(2))) float v2f;
typedef __attribute__((ext_vector_type(8))) float v8f;

// gfx1250 async global->LDS copy (ASYNCcnt-tracked), 16 bytes per lane issue.
// Toolchain signature: (v4i addrspace(1)*, v4i addrspace(3)*, imm offset, imm cpol)
typedef int v4i_vs __attribute__((vector_size(16)));
typedef __attribute__((address_space(1))) v4i_vs as1_v4i;
typedef __attribute__((address_space(3))) v4i_vs as3_v4i;

__device__ __forceinline__ void async_copy_b128(const void* g, void* l) {
  __builtin_amdgcn_global_load_async_to_lds_b128(
      (as1_v4i*)g, (as3_v4i*)l, /*imm offset=*/0, /*cpol=*/0);
}

// ---------------------------------------------------------------------------
// 1) init: zero agg & out, deg = 1.0 (self loop)
// ---------------------------------------------------------------------------
__global__ void gcn_init(float* __restrict__ agg, float* __restrict__ out,
                         float* __restrict__ deg, int N) {
  int idx = blockIdx.x * blockDim.x + threadIdx.x;
  int tot = N * 64;
  if (idx < tot) { agg[idx] = 0.0f; out[idx] = 0.0f; }
  if (idx < N)   deg[idx] = 1.0f;
}

// ---------------------------------------------------------------------------
// 2) degree over dst (fp32 atomics resolve in L2; deg buffer is 400 KB)
// ---------------------------------------------------------------------------
__global__ void gcn_degree(const int* __restrict__ dst, float* __restrict__ deg, int E) {
  int e = blockIdx.x * blockDim.x + threadIdx.x;
  if (e < E)
    __hip_atomic_fetch_add(&deg[dst[e]], 1.0f, __ATOMIC_RELAXED,
                           __HIP_MEMORY_SCOPE_AGENT);
}

// 3) dis = rsqrt(deg) in place (deg >= 1 always)
__global__ void gcn_rsqrt(float* __restrict__ deg, int N) {
  int i = blockIdx.x * blockDim.x + threadIdx.x;
  if (i < N) deg[i] = rsqrtf(deg[i]);
}

// ---------------------------------------------------------------------------
// 4) fp32 WMMA GEMM: out[N x 64] = in[N x K] @ W[K x 64]
//    grid.x = N/16 M-tiles, 128 threads = 4 waves, wave w -> cols [16w,16w+16)
//    Both the 16xK A tile and the Kx64 W matrix are staged into LDS with the
//    gfx1250 async copy engine, then the inner loop is pure LDS + WMMA:
//    K/4 chained V_WMMA_F32_16X16X4_F32 per wave.
// ---------------------------------------------------------------------------
template <int K>
__global__ void gcn_gemm_wmma(const float* __restrict__ in, const float* __restrict__ W,
                              float* __restrict__ out) {
  __shared__ float As[16 * K];   // A tile, row-major [16][K]
  __shared__ float Ws[K * 64];   // W,      row-major [K][64]

  const int m0 = blockIdx.x * 16;

  // ---- async bulk staging (ASYNCcnt) ----
  {
    const char* ga = (const char*)(in + (size_t)m0 * K);  // 16 contiguous rows
    char*       la = (char*)As;
    for (int t = threadIdx.x * 16; t < 16 * K * 4; t += blockDim.x * 16)
      async_copy_b128(ga + t, la + t);

    const char* gw = (const char*)W;
    char*       lw = (char*)Ws;
    for (int t = threadIdx.x * 16; t < K * 64 * 4; t += blockDim.x * 16)
      async_copy_b128(gw + t, lw + t);

    __builtin_amdgcn_s_wait_asynccnt(0);
  }
  __syncthreads();

  const int lane = threadIdx.x & 31;
  const int wave = threadIdx.x >> 5;   // 4 waves cover the 64 output columns
  const int half = lane >> 4;          // lanes 16-31 hold the upper K pair
  const int l16  = lane & 15;
  const int n0   = wave * 16;

  v8f acc = {};
#pragma unroll
  for (int s = 0; s < K / 4; ++s) {
    const int ka = s * 4 + half * 2;
    // A 16x4: lane (l16, half) holds in[m0+l16][ka], in[m0+l16][ka+1]
    v2f a = *(const v2f*)(&As[l16 * K + ka]);
    // B 4x16: VGPR0 = W[ka][n0+l16], VGPR1 = W[ka+1][n0+l16]
    v2f b;
    b.x = Ws[(ka + 0) * 64 + n0 + l16];
    b.y = Ws[(ka + 1) * 64 + n0 + l16];
    acc = __builtin_amdgcn_wmma_f32_16x16x4_f32(
        /*neg_a=*/false, a, /*neg_b=*/false, b,
        /*c_mod=*/(short)0, acc, /*reuse_a=*/false, /*reuse_b=*/false);
  }

  // C/D layout: VGPR r -> row m0 + r + 8*half, col n0 + l16
#pragma unroll
  for (int r = 0; r < 8; ++r)
    out[(size_t)(m0 + half * 8 + r) * 64 + n0 + l16] = acc[r];
}

// ---------------------------------------------------------------------------
// 5) edge gather/scatter: one wave per edge, lane covers 2 of 64 features.
//    agg[dst] += dis[src]*dis[dst] * h[src]   (L2-resident fp32 atomics)
// ---------------------------------------------------------------------------
__global__ void gcn_scatter(const int* __restrict__ src, const int* __restrict__ dst,
                            const float* __restrict__ dis, const float* __restrict__ h,
                            float* __restrict__ agg, int E) {
  long long gid = (long long)blockIdx.x * blockDim.x + threadIdx.x;
  int e    = (int)(gid >> 5);
  int lane = (int)(gid & 31);
  if (e >= E) return;
  int s = src[e], d = dst[e];
  float w = dis[s] * dis[d];
  v2f v = *(const v2f*)(h + (size_t)s * 64 + lane * 2);
  float* p = agg + (size_t)d * 64 + lane * 2;
  __hip_atomic_fetch_add(p,     v.x * w, __ATOMIC_RELAXED, __HIP_MEMORY_SCOPE_AGENT);
  __hip_atomic_fetch_add(p + 1, v.y * w, __ATOMIC_RELAXED, __HIP_MEMORY_SCOPE_AGENT);
}

// 6) h = relu(agg + dis^2 * h1 + b)   (self-loop folded in, in place on agg)
__global__ void gcn_bias_relu(const float* __restrict__ h1, const float* __restrict__ dis,
                              const float* __restrict__ b, float* __restrict__ agg, int N) {
  int idx = blockIdx.x * blockDim.x + threadIdx.x;
  if (idx >= N * 64) return;
  int i = idx >> 6, c = idx & 63;
  float di = dis[i];
  float v = agg[idx] + di * di * h1[idx] + b[c];
  agg[idx] = v > 0.0f ? v : 0.0f;
}

// ---------------------------------------------------------------------------
// 7) out = log_softmax(out + dis^2 * h2 + b2), one wave32 per node,
//    lane holds 2 of 64 logits; butterfly-shuffle max & sum reductions.
// ---------------------------------------------------------------------------
__global__ void gcn_logsoftmax(const float* __restrict__ h2, const float* __restrict__ dis,
                               const float* __restrict__ b, float* __restrict__ out, int N) {
  long long gid = (long long)blockIdx.x * blockDim.x + threadIdx.x;
  int node = (int)(gid >> 5);
  int lane = (int)(gid & 31);
  if (node >= N) return;
  float di  = dis[node];
  float di2 = di * di;
  size_t base = (size_t)node * 64 + lane * 2;
  float vx = out[base]     + di2 * h2[base]     + b[lane * 2];
  float vy = out[base + 1] + di2 * h2[base + 1] + b[lane * 2 + 1];

  float m = fmaxf(vx, vy);
#pragma unroll
  for (int off = 16; off; off >>= 1) m = fmaxf(m, __shfl_xor(m, off, 32));
  float s = __expf(vx - m) + __expf(vy - m);
#pragma unroll
  for (int off = 16; off; off >>= 1) s += __shfl_xor(s, off, 32);
  float lse = m + __logf(s);

  out[base]     = vx - lse;
  out[base + 1] = vy - lse;
}

// ---------------------------------------------------------------------------
extern "C" void kernel_launch(void* const* d_in, const int* in_sizes, int n_in,
                              void* d_out, int out_size, void* d_ws, size_t ws_size,
                              hipStream_t stream) {
  const float* x  = (const float*)d_in[0];   // [N,128]
  const int*   ei = (const int*)d_in[1];     // [2,E]
  const float* W1 = (const float*)d_in[2];   // [128,64]
  const float* b1 = (const float*)d_in[3];   // [64]
  const float* W2 = (const float*)d_in[4];   // [64,64]
  const float* b2 = (const float*)d_in[5];   // [64]
  float* out = (float*)d_out;                // [N,64]

  const int N = in_sizes[0] / 128;           // 100000 (multiple of 16)
  const int E = in_sizes[1] / 2;             // 1600000
  const int* src = ei;
  const int* dst = ei + E;

  // workspace: dis[N] | h1[N*64] | agg[N*64]  (~52 MB)
  float* ws  = (float*)d_ws;
  float* dis = ws;
  float* h1  = ws + N;
  float* agg = h1 + (size_t)N * 64;

  const int T = 256;
  const int featBlocks = (N * 64 + T - 1) / T;
  const int edgeWaveBlocks = (int)(((long long)E * 32 + T - 1) / T);

  gcn_init<<<featBlocks, T, 0, stream>>>(agg, out, dis, N);
  gcn_degree<<<(E + T - 1) / T, T, 0, stream>>>(dst, dis, E);
  gcn_rsqrt<<<(N + T - 1) / T, T, 0, stream>>>(dis, N);

  // layer 1
  gcn_gemm_wmma<128><<<N / 16, 128, 0, stream>>>(x, W1, h1);
  gcn_scatter<<<edgeWaveBlocks, T, 0, stream>>>(src, dst, dis, h1, agg, E);
  gcn_bias_relu<<<featBlocks, T, 0, stream>>>(h1, dis, b1, agg, N);

  // layer 2 (h2 reuses h1 buffer; scatter accumulates straight into d_out)
  gcn_gemm_wmma<64><<<N / 16, 128, 0, stream>>>(agg, W2, h1);
  gcn_scatter<<<edgeWaveBlocks, T, 0, stream>>>(src, dst, dis, h1, out, E);
  gcn_logsoftmax<<<(int)(((long long)N * 32 + T - 1) / T), T, 0, stream>>>(h1, dis, b2, out, N);
}